// HierarchicalGraphPooling_34127810134470
// MI455X (gfx1250) — compile-verified
//
#include <hip/hip_runtime.h>

#define DFEAT 256

typedef float v2f __attribute__((ext_vector_type(2)));
typedef float v4f __attribute__((ext_vector_type(4)));
typedef float v8f __attribute__((ext_vector_type(8)));

// ---------------- small utility kernels ----------------

__global__ void fill_deg_kernel(float* deg, int n) {
  int i = blockIdx.x * blockDim.x + threadIdx.x;
  if (i < n) deg[i] = 1.0f;   // self-loop contribution
}

__global__ void fill_inv_kernel(int* inv, int n, int k) {
  int i = blockIdx.x * blockDim.x + threadIdx.x;
  if (i < n) inv[i] = k;      // "removed" marker
}

__global__ void zero_kernel(float* p, int count) {
  int i = blockIdx.x * blockDim.x + threadIdx.x;
  if (i < count) p[i] = 0.0f;
}

__global__ void convert_edges_kernel(const int* __restrict__ ei, int E,
                                     int* __restrict__ src, int* __restrict__ dst,
                                     int* __restrict__ mask) {
  int e = blockIdx.x * blockDim.x + threadIdx.x;
  if (e < E) { src[e] = ei[e]; dst[e] = ei[E + e]; mask[e] = 1; }
}

__global__ void scatter_deg_kernel(const int* __restrict__ dst, const int* __restrict__ mask,
                                   float* __restrict__ deg, int E) {
  int e = blockIdx.x * blockDim.x + threadIdx.x;
  if (e < E && mask[e]) atomicAdd(&deg[dst[e]], 1.0f);
}

__global__ void dinv_kernel(const float* __restrict__ deg, float* __restrict__ dinv, int n) {
  int i = blockIdx.x * blockDim.x + threadIdx.x;
  if (i < n) dinv[i] = rsqrtf(deg[i]);
}

// ---------------- WMMA FP32 GEMM: H = X @ W  (X: M x 256, W: 256 x 256) ----------------
// Workgroup = 8 waves sharing one N-tile (blockIdx.y). The 256x16 W panel is staged
// once into LDS with coalesced b128 loads; each wave then computes one 16x16 output
// tile via V_WMMA_F32_16X16X4_F32, K stepped by 4, B fragments fed from LDS.
// A frag (16x4 f32, 2 VGPR/lane): lane l -> M = l&15, holds K = 2*(l>>4) + {0,1}
// B frag (4x16 f32, 2 VGPR/lane): lane l -> N = l&15, holds K = 2*(l>>4) + {0,1}
// C/D (16x16 f32, 8 VGPR): VGPR j, lane l -> M = j + 8*(l>>4), N = l&15
//
// Ragged M (level 3 only): A rows are clamped so loads stay in-bounds, and the
// destination buffer is guaranteed to have capacity for mtiles*16 rows, so the
// epilogue stores padded rows unconditionally (no per-row EXEC toggling).

__global__ void __launch_bounds__(256) gemm_wmma_kernel(const float* __restrict__ X,
                                                        const float* __restrict__ W,
                                                        float* __restrict__ H, int M) {
  __shared__ float ldsW[DFEAT * 16];   // W[:, tn*16 .. tn*16+16), layout [k][n], 16 KB

  const int tid  = threadIdx.x;
  const int lane = tid & 31;
  const int wave = tid >> 5;
  const int tn   = blockIdx.y;
  const int tm   = blockIdx.x * 8 + wave;
  const int mtiles = (M + 15) >> 4;

  // cooperative staged load of the W panel (1024 float4s, 64B-contiguous groups)
  {
    const float* Wp = W + tn * 16;
#pragma unroll
    for (int c = 0; c < 4; ++c) {
      int vec = c * 256 + tid;
      int k = vec >> 2;
      int n4 = (vec & 3) * 4;
      v4f v = *(const v4f*)(Wp + (long)k * DFEAT + n4);
      *(v4f*)(ldsW + k * 16 + n4) = v;
    }
  }
  __syncthreads();

  const int mr = lane & 15;
  const int hi = lane >> 4;
  const int row  = tm * 16 + mr;
  const int rowc = (row < M) ? row : (M - 1);  // clamp keeps padded-tile loads valid
  const int col  = tn * 16 + mr;

  if (tm < mtiles) {                           // wave-uniform: EXEC all-ones at WMMA
    const float* Xr = X + (long)rowc * DFEAT + 2 * hi;
    v8f acc = {};
#pragma unroll 8
    for (int k = 0; k < DFEAT; k += 4) {
      v2f a = *(const v2f*)(Xr + k);
      v2f b;
      b.x = ldsW[(k + 2 * hi) * 16 + mr];
      b.y = ldsW[(k + 2 * hi + 1) * 16 + mr];
      acc = __builtin_amdgcn_wmma_f32_16x16x4_f32(false, a, false, b, (short)0, acc,
                                                  false, false);
    }
    union { v8f v; float f[8]; } u;
    u.v = acc;
    float* Hr = H + (long)(tm * 16 + 8 * hi) * DFEAT + col;
#pragma unroll
    for (int j = 0; j < 8; ++j) {
      Hr[(long)j * DFEAT] = u.f[j];            // unconditional: dest has padded capacity
    }
  }
}

// ---------------- edge feature aggregation: agg[dst] += h[src] * dinv[src]*dinv[dst] ----------------

__global__ void scatter_feat_kernel(const float* __restrict__ h, const int* __restrict__ src,
                                    const int* __restrict__ dst, const int* __restrict__ mask,
                                    const float* __restrict__ dinv, float* __restrict__ agg,
                                    int E) {
  int id = blockIdx.x * blockDim.x + threadIdx.x;   // one thread per (edge, feature)
  int e = id >> 8;
  int f = id & 255;
  if (e >= E) return;
  if (!mask[e]) return;
  int s = src[e], d = dst[e];
  float c = dinv[s] * dinv[d];
  atomicAdd(&agg[(long)d * DFEAT + f], h[(long)s * DFEAT + f] * c);
}

// ---------------- epilogue: h2 = ELU(agg + h*dinv^2 + b)   (in-place into agg) ----------------

__global__ void finish_elu_kernel(const float* __restrict__ h, const float* __restrict__ dinv,
                                  const float* __restrict__ b, float* __restrict__ agg, int n) {
  int id = blockIdx.x * blockDim.x + threadIdx.x;
  if (id >= n * DFEAT) return;
  int row = id >> 8;
  int f = id & 255;
  float di = dinv[row];
  float v = agg[id] + h[id] * di * di + b[f];
  agg[id] = (v > 0.0f) ? v : expm1f(v);
}

// ---------------- score = (h2 @ p) / ||p||   (one block per row) ----------------

__global__ void __launch_bounds__(256) score_kernel(const float* __restrict__ h2,
                                                    const float* __restrict__ p,
                                                    float* __restrict__ score, int n) {
  int row = blockIdx.x;
  int f = threadIdx.x;
  __shared__ float sdot[256];
  __shared__ float spp[256];
  float pv = p[f];
  sdot[f] = h2[(long)row * DFEAT + f] * pv;
  spp[f] = pv * pv;
  __syncthreads();
  for (int s = 128; s > 0; s >>= 1) {
    if (f < s) { sdot[f] += sdot[f + s]; spp[f] += spp[f + s]; }
    __syncthreads();
  }
  if (f == 0) score[row] = sdot[0] / sqrtf(spp[0]);
}

// ---------------- float -> order-preserving uint keys ----------------

__global__ void keys_kernel(const float* __restrict__ score, unsigned* __restrict__ keys, int n) {
  int i = blockIdx.x * blockDim.x + threadIdx.x;
  if (i >= n) return;
  unsigned u = __float_as_uint(score[i]);
  u = (u & 0x80000000u) ? ~u : (u | 0x80000000u);
  keys[i] = u;
}

// ---------------- deterministic single-block radix top-k selection ----------------
// Finds the k-th largest key via 32-bit radix select, then compacts the selected
// node ids into perm[0..k) in index order (deterministic, tie-safe) and fills inv.

__global__ void __launch_bounds__(1024) select_topk_kernel(const unsigned* __restrict__ keys,
                                                           int n, int k,
                                                           int* __restrict__ perm,
                                                           int* __restrict__ inv) {
  const int tid = threadIdx.x;
  const int NT = 1024;
  __shared__ int lds[1024];

  // ---- radix select threshold T = k-th largest key ----
  unsigned prefix = 0u;
  int kr = k;
  for (int bit = 31; bit >= 0; --bit) {
    unsigned himask = (bit == 31) ? 0u : (0xFFFFFFFFu << (bit + 1));
    int cnt = 0;
    for (int i = tid; i < n; i += NT) {
      unsigned kk = keys[i];
      if ((kk & himask) == prefix && ((kk >> bit) & 1u)) cnt++;
    }
    lds[tid] = cnt; __syncthreads();
    for (int s = NT / 2; s > 0; s >>= 1) {
      if (tid < s) lds[tid] += lds[tid + s];
      __syncthreads();
    }
    int total = lds[0]; __syncthreads();
    if (total >= kr) prefix |= (1u << bit);
    else kr -= total;
  }
  const unsigned T = prefix;

  // ---- count strictly greater than T ----
  int cg = 0;
  for (int i = tid; i < n; i += NT) if (keys[i] > T) cg++;
  lds[tid] = cg; __syncthreads();
  for (int s = NT / 2; s > 0; s >>= 1) {
    if (tid < s) lds[tid] += lds[tid + s];
    __syncthreads();
  }
  const int cntG = lds[0]; __syncthreads();
  const int need = k - cntG;   // how many ==T elements to take (index order)

  // ---- ordered compaction via block-wide scans ----
  int takenG = 0, takenE = 0;
  for (int base = 0; base < n; base += NT) {
    int i = base + tid;
    unsigned kk = (i < n) ? keys[i] : 0u;
    int g = (i < n && kk > T) ? 1 : 0;
    int e = (i < n && kk == T) ? 1 : 0;

    // inclusive scan of g
    lds[tid] = g; __syncthreads();
    for (int off = 1; off < NT; off <<= 1) {
      int t = (tid >= off) ? lds[tid - off] : 0;
      __syncthreads();
      lds[tid] += t;
      __syncthreads();
    }
    int sg = lds[tid];
    int totG = lds[NT - 1];
    __syncthreads();

    // inclusive scan of e
    lds[tid] = e; __syncthreads();
    for (int off = 1; off < NT; off <<= 1) {
      int t = (tid >= off) ? lds[tid - off] : 0;
      __syncthreads();
      lds[tid] += t;
      __syncthreads();
    }
    int se = lds[tid];
    int totE = lds[NT - 1];
    __syncthreads();

    if (g) {
      int slot = takenG + sg - 1;
      perm[slot] = i;
      inv[i] = slot;
    }
    if (e) {
      int r = takenE + se - 1;
      if (r < need) {
        int slot = cntG + r;
        perm[slot] = i;
        inv[i] = slot;
      }
    }
    takenG += totG;
    takenE += totE;
  }
}

// ---------------- xp[i] = h2[perm[i]] * tanh(score[perm[i]]) ----------------

__global__ void gather_scale_kernel(const float* __restrict__ h2, const float* __restrict__ score,
                                    const int* __restrict__ perm, float* __restrict__ xp, int k) {
  int i = blockIdx.x;
  int f = threadIdx.x;
  int node = perm[i];
  float t = tanhf(score[node]);
  xp[(long)i * DFEAT + f] = h2[(long)node * DFEAT + f] * t;
}

// ---------------- edge relabel: ns = inv[src], keep if both < k ----------------

__global__ void relabel_kernel(const int* __restrict__ osrc, const int* __restrict__ odst,
                               const int* __restrict__ omask, const int* __restrict__ inv,
                               int k, int E, int* __restrict__ nsrc, int* __restrict__ ndst,
                               int* __restrict__ nmask) {
  int e = blockIdx.x * blockDim.x + threadIdx.x;
  if (e >= E) return;
  int ns = inv[osrc[e]];
  int nd = inv[odst[e]];
  int m = (omask[e] && ns < k && nd < k) ? 1 : 0;
  nsrc[e] = m ? ns : 0;
  ndst[e] = m ? nd : 0;
  nmask[e] = m;
}

// ---------------- column mean over k rows -> out[256] ----------------

__global__ void __launch_bounds__(256) colmean_kernel(const float* __restrict__ xp, int k,
                                                      float* __restrict__ out) {
  int f = blockIdx.x;     // column
  int t = threadIdx.x;
  float s = 0.0f;
  for (int i = t; i < k; i += 256) s += xp[(long)i * DFEAT + f];
  __shared__ float red[256];
  red[t] = s; __syncthreads();
  for (int w = 128; w > 0; w >>= 1) {
    if (t < w) red[t] += red[t + w];
    __syncthreads();
  }
  if (t == 0) out[f] = red[0] / (float)k;
}

// ---------------- host-side level orchestration ----------------

static void run_level(const float* x, int n, int k,
                      const int* esrc, const int* edst, const int* emask,
                      int* nsrc, int* ndst, int* nmask,
                      const float* W, const float* b, const float* p,
                      float* hbuf, float* aggbuf, float* xpOut,
                      float* deg, float* dinv, float* score, unsigned* keys,
                      int* perm, int* inv, float* outRepr, hipStream_t stream) {
  const int E_ = 320000;
  fill_deg_kernel<<<(n + 255) / 256, 256, 0, stream>>>(deg, n);
  scatter_deg_kernel<<<(E_ + 255) / 256, 256, 0, stream>>>(edst, emask, deg, E_);
  dinv_kernel<<<(n + 255) / 256, 256, 0, stream>>>(deg, dinv, n);

  int mtiles = (n + 15) / 16;
  dim3 ggrid((mtiles + 7) / 8, DFEAT / 16);
  gemm_wmma_kernel<<<ggrid, 256, 0, stream>>>(x, W, hbuf, n);

  int nelem = n * DFEAT;
  zero_kernel<<<(nelem + 255) / 256, 256, 0, stream>>>(aggbuf, nelem);
  scatter_feat_kernel<<<E_, 256, 0, stream>>>(hbuf, esrc, edst, emask, dinv, aggbuf, E_);
  finish_elu_kernel<<<(nelem + 255) / 256, 256, 0, stream>>>(hbuf, dinv, b, aggbuf, n);

  score_kernel<<<n, 256, 0, stream>>>(aggbuf, p, score, n);
  keys_kernel<<<(n + 255) / 256, 256, 0, stream>>>(score, keys, n);
  fill_inv_kernel<<<(n + 255) / 256, 256, 0, stream>>>(inv, n, k);
  select_topk_kernel<<<1, 1024, 0, stream>>>(keys, n, k, perm, inv);

  gather_scale_kernel<<<k, 256, 0, stream>>>(aggbuf, score, perm, xpOut, k);
  relabel_kernel<<<(E_ + 255) / 256, 256, 0, stream>>>(esrc, edst, emask, inv, k, E_,
                                                       nsrc, ndst, nmask);
  colmean_kernel<<<DFEAT, 256, 0, stream>>>(xpOut, k, outRepr);
}

extern "C" void kernel_launch(void* const* d_in, const int* in_sizes, int n_in,
                              void* d_out, int out_size, void* d_ws, size_t ws_size,
                              hipStream_t stream) {
  (void)in_sizes; (void)n_in; (void)out_size; (void)ws_size;
  const int N_ = 20000, E_ = 320000;
  const int K1 = 10000, K2 = 5000, K3 = 2500;

  const float* x  = (const float*)d_in[0];
  const int*   ei = (const int*)d_in[1];
  const float* W1 = (const float*)d_in[2];
  const float* b1 = (const float*)d_in[3];
  const float* p1 = (const float*)d_in[4];
  const float* W2 = (const float*)d_in[5];
  const float* b2 = (const float*)d_in[6];
  const float* p2 = (const float*)d_in[7];
  const float* W3 = (const float*)d_in[8];
  const float* b3 = (const float*)d_in[9];
  const float* p3 = (const float*)d_in[10];
  float* out = (float*)d_out;

  // workspace carve-up (~59 MB)
  float* bufH   = (float*)d_ws;                         // N*256 f32 (h / xp2)
  float* bufAgg = bufH + (size_t)N_ * DFEAT;            // N*256 f32 (agg / h2)
  float* bufC   = bufAgg + (size_t)N_ * DFEAT;          // K1*256 f32 (xp1 / h3 / xp3)
  float* deg    = bufC + (size_t)K1 * DFEAT;
  float* dinv   = deg + N_;
  float* score  = dinv + N_;
  unsigned* keys = (unsigned*)(score + N_);
  int* perm  = (int*)(keys + N_);                       // K1 ints
  int* inv   = perm + K1;                               // N ints
  int* srcA  = inv + N_;
  int* dstA  = srcA + E_;
  int* maskA = dstA + E_;
  int* srcB  = maskA + E_;
  int* dstB  = srcB + E_;
  int* maskB = dstB + E_;

  convert_edges_kernel<<<(E_ + 255) / 256, 256, 0, stream>>>(ei, E_, srcA, dstA, maskA);

  // level 1: x(N) -> xp1(K1) in bufC, fine repr -> out[0:256)
  run_level(x, N_, K1, srcA, dstA, maskA, srcB, dstB, maskB, W1, b1, p1,
            bufH, bufAgg, bufC, deg, dinv, score, keys, perm, inv, out + 0, stream);
  // level 2: xp1(bufC) -> xp2(K2) in bufH, mid repr -> out[256:512)
  run_level(bufC, K1, K2, srcB, dstB, maskB, srcA, dstA, maskA, W2, b2, p2,
            bufH, bufAgg, bufH, deg, dinv, score, keys, perm, inv, out + 256, stream);
  // level 3: xp2(bufH) -> xp3(K3) in bufC (capacity K1 rows covers the 5008-row
  // padded GEMM output), coarse repr -> out[512:768)
  run_level(bufH, K2, K3, srcA, dstA, maskA, srcB, dstB, maskB, W3, b3, p3,
            bufC, bufAgg, bufC, deg, dinv, score, keys, perm, inv, out + 512, stream);
}